// LieSE3_37065567764674
// MI455X (gfx1250) — compile-verified
//
#include <hip/hip_runtime.h>
#include <hip/hip_bf16.h>

#define TRANSLATION_SCALE 1.0f
#define ROTATION_SCALE    0.1f
#define LIE_EPS           1e-05f

typedef float v4f __attribute__((ext_vector_type(4)));
typedef int   v4i __attribute__((ext_vector_type(4)));

#ifndef __has_builtin
#define __has_builtin(x) 0
#endif

__device__ __forceinline__ void wait_asynccnt0() {
#if __has_builtin(__builtin_amdgcn_s_wait_asynccnt)
    __builtin_amdgcn_s_wait_asynccnt(0);
#else
    asm volatile("s_wait_asynccnt 0" ::: "memory");
#endif
}

// One block = 256 elements (8 wave32s).
// Stage the block's 256*3 floats (3072 B) into LDS with 192 async B128 DMAs,
// then each thread computes one 4x4 SE3 matrix and streams it out with
// four non-temporal B128 stores. All indexing is 32-bit (max input float
// index 6e6, max output byte offset 128 MB — both << 2^31).
__global__ __launch_bounds__(256) void lie_se3_kernel(
    const float* __restrict__ uv, float* __restrict__ out, int B) {
    __shared__ float s_in[256 * 3];

    const int tid         = threadIdx.x;
    const int base        = (int)blockIdx.x * 256;   // first element of block
    const int totalFloats = B * 3;                   // fits in int (6e6)
    const int gfloat0     = base * 3;

    // ---- cooperative async staging: global -> LDS (B128 per lane) ----
    if (tid < 192) {
        const int chunk = gfloat0 + tid * 4;         // float index of this 16B chunk
        if (chunk + 4 <= totalFloats) {
#if __has_builtin(__builtin_amdgcn_global_load_async_to_lds_b128)
            typedef __attribute__((address_space(1))) v4i gv4i_t;   // global (AS1)
            typedef __attribute__((address_space(3))) v4i lv4i_t;   // LDS (AS3)
            __builtin_amdgcn_global_load_async_to_lds_b128(
                (gv4i_t*)(uv + chunk), (lv4i_t*)(&s_in[tid * 4]),
                /*offset=*/0, /*cpol=*/0);
#else
            *(v4f*)(&s_in[tid * 4]) = *(const v4f*)(uv + chunk);
#endif
        } else {
            // ragged final chunk (only if B*3 % 4 != 0): scalar guarded copy
            for (int k = 0; k < 4; ++k) {
                const int f = chunk + k;
                if (f < totalFloats) s_in[tid * 4 + k] = uv[f];
            }
        }
    }
    wait_asynccnt0();
    __syncthreads();

    const int i = base + tid;
    if (i >= B) return;

    // ---- per-element math (LDS reads: stride 3 is coprime with 64 banks) ----
    const float x  = s_in[tid * 3 + 0] * TRANSLATION_SCALE;
    const float y  = s_in[tid * 3 + 1] * TRANSLATION_SCALE;
    const float w  = s_in[tid * 3 + 2] * ROTATION_SCALE;

    const float t2    = w * w;
    const float theta = __builtin_fabsf(w);              // sqrt(w*w)
    const float invT  = __builtin_amdgcn_rcpf(theta + LIE_EPS);
    const float invT2 = __builtin_amdgcn_rcpf(t2 + LIE_EPS);

    const float s  = __sinf(theta);
    const float c  = __cosf(theta);
    const float A  = s * invT;                 // sin(t)/t
    const float Bc = (1.0f - c) * invT2;       // (1-cos)/t^2
    const float C  = (1.0f - A) * invT2;       // (1-A)/t^2

    // R = I + A*skew + Bc*skew2  (z-only rotation)
    const float R00 = 1.0f - Bc * t2;          // also R11
    const float Aw  = A * w;                   // R10 = Aw, R01 = -Aw
    // V = I + Bc*skew + C*skew2
    const float V00 = 1.0f - C * t2;           // also V11
    const float Bw  = Bc * w;                  // V10 = Bw, V01 = -Bw

    // t = V * (x, y, 0)
    const float t0 = V00 * x - Bw * y;
    const float t1 = Bw * x + V00 * y;

    // ---- stream out 4x4 row-major with non-temporal B128 stores ----
    v4f* o = (v4f*)(out + i * 16);             // 32-bit scaled offset
    const v4f r0 = { R00,  -Aw, 0.0f,  t0  };
    const v4f r1 = { Aw,   R00, 0.0f,  t1  };
    const v4f r2 = { 0.0f, 0.0f, 1.0f, 0.0f };
    const v4f r3 = { 0.0f, 0.0f, 0.0f, 1.0f };
    __builtin_nontemporal_store(r0, o + 0);
    __builtin_nontemporal_store(r1, o + 1);
    __builtin_nontemporal_store(r2, o + 2);
    __builtin_nontemporal_store(r3, o + 3);
}

extern "C" void kernel_launch(void* const* d_in, const int* in_sizes, int n_in,
                              void* d_out, int out_size, void* d_ws, size_t ws_size,
                              hipStream_t stream) {
    const float* uv  = (const float*)d_in[0];
    float*       out = (float*)d_out;
    const int B = in_sizes[0] / 3;              // 2,000,000
    const int blocks = (B + 255) / 256;
    hipLaunchKernelGGL(lie_se3_kernel, dim3(blocks), dim3(256), 0, stream,
                       uv, out, B);
}